// DenseFlashAttention_19859928777368
// MI455X (gfx1250) — compile-verified
//
#include <hip/hip_runtime.h>
#include <hip/hip_bf16.h>

#define N_NODES 10000
#define N_EDGES 160000
#define FDIM 64
#define NH 4
#define MTILES 625   // N_NODES / 16 exactly

typedef float v2f __attribute__((ext_vector_type(2)));
typedef float v8f __attribute__((ext_vector_type(8)));

__device__ __forceinline__ float softplusf(float v) {
  return v > 20.0f ? v : log1pf(__expf(v));
}

// monotone float <-> uint mapping for atomic max on floats
__device__ __forceinline__ unsigned fenc(float f) {
  unsigned u = __float_as_uint(f);
  return (u & 0x80000000u) ? ~u : (u | 0x80000000u);
}
__device__ __forceinline__ float fdec(unsigned k) {
  return (k & 0x80000000u) ? __uint_as_float(k & 0x7FFFFFFFu) : __uint_as_float(~k);
}
#define NEG_INF_KEY 0x007FFFFFu   // fenc(-inf)

// ---------------------------------------------------------------- init
__global__ void ega_init(unsigned* m_r, unsigned* m_t, float* d_r, float* d_t,
                         float* msg) {
  int i = blockIdx.x * blockDim.x + threadIdx.x;
  if (i < N_NODES * NH) {
    m_r[i] = NEG_INF_KEY;
    m_t[i] = NEG_INF_KEY;
    d_r[i] = 0.0f;
    d_t[i] = 0.0f;
  }
  if (i < N_NODES * FDIM) msg[i] = 0.0f;
}

// ------------------------------------------------- WMMA projections
// tasks: mat in {0=energy,1=radial,2=tang} x head x mtile, one wave each.
// Each wave computes a 16x64 tile of out[h] = x @ W[h] with V_WMMA_F32_16X16X4_F32.
// mat==0 additionally reduces against radial/tangential score vectors to
// produce per-node logit scalars s_r[n][h], s_t[n][h] (instead of storing energy).
__global__ void ega_proj_wmma(const float* __restrict__ x,
                              const float* __restrict__ wp,
                              const float* __restrict__ wr,
                              const float* __restrict__ wt,
                              const float* __restrict__ rs,
                              const float* __restrict__ ts,
                              float* __restrict__ radial_tab,
                              float* __restrict__ tang_tab,
                              float* __restrict__ s_r,
                              float* __restrict__ s_t) {
  int wave = (blockIdx.x * blockDim.x + threadIdx.x) >> 5;
  if (wave >= 3 * NH * MTILES) return;
  int mtile = wave % MTILES;
  int rest  = wave / MTILES;
  int head  = rest & (NH - 1);
  int mat   = rest >> 2;

  int lane = threadIdx.x & 31;
  int half = lane >> 4;      // 0: lanes 0-15, 1: lanes 16-31
  int lm   = lane & 15;
  int m0   = mtile * 16;

  const float* W = ((mat == 0) ? wp : (mat == 1) ? wr : wt) + head * FDIM * FDIM;

  // Preload A (16 rows x 64 K) : lane lm holds row m0+lm; v0/v1 = K pair at half*2.
  v2f areg[16];
  const float* xrow = x + (size_t)(m0 + lm) * FDIM + half * 2;
#pragma unroll
  for (int k4 = 0; k4 < 16; ++k4) areg[k4] = *(const v2f*)(xrow + k4 * 4);

  v8f acc[4];
#pragma unroll
  for (int t = 0; t < 4; ++t) acc[t] = (v8f){0.f, 0.f, 0.f, 0.f, 0.f, 0.f, 0.f, 0.f};

#pragma unroll
  for (int t = 0; t < 4; ++t) {
#pragma unroll
    for (int k4 = 0; k4 < 16; ++k4) {
      int kk = k4 * 4 + half * 2;   // B VGPR0: rows K, K+2 ; VGPR1: rows K+1, K+3
      v2f b;
      b.x = W[kk * FDIM + t * 16 + lm];
      b.y = W[(kk + 1) * FDIM + t * 16 + lm];
      acc[t] = __builtin_amdgcn_wmma_f32_16x16x4_f32(
          false, areg[k4], false, b, (short)0, acc[t], false, false);
    }
  }

  if (mat == 0) {
    // per-row dot with score vectors; C/D element (M = v + 8*half, Ncol = 16t+lm)
    float pr[8], pt[8];
#pragma unroll
    for (int v = 0; v < 8; ++v) { pr[v] = 0.f; pt[v] = 0.f; }
#pragma unroll
    for (int t = 0; t < 4; ++t) {
      float rsv = rs[head * FDIM + t * 16 + lm];
      float tsv = ts[head * FDIM + t * 16 + lm];
#pragma unroll
      for (int v = 0; v < 8; ++v) {
        pr[v] += acc[t][v] * rsv;
        pt[v] += acc[t][v] * tsv;
      }
    }
#pragma unroll
    for (int v = 0; v < 8; ++v) {
#pragma unroll
      for (int m = 1; m < 16; m <<= 1) {
        pr[v] += __shfl_xor(pr[v], m, 32);
        pt[v] += __shfl_xor(pt[v], m, 32);
      }
    }
    if (lm == 0) {
#pragma unroll
      for (int v = 0; v < 8; ++v) {
        int row = m0 + v + half * 8;
        s_r[row * NH + head] = pr[v];
        s_t[row * NH + head] = pt[v];
      }
    }
  } else {
    float* tab = ((mat == 1) ? radial_tab : tang_tab) + (size_t)head * N_NODES * FDIM;
#pragma unroll
    for (int t = 0; t < 4; ++t) {
#pragma unroll
      for (int v = 0; v < 8; ++v) {
        int row = m0 + v + half * 8;
        tab[(size_t)row * FDIM + t * 16 + lm] = acc[t][v];
      }
    }
  }
}

// ------------------------------------------------- edge logits + segment max
__global__ void ega_edge_logits(const int* __restrict__ ei,
                                const float* __restrict__ elen,
                                const float* __restrict__ s_r,
                                const float* __restrict__ s_t,
                                const float* __restrict__ p_ls,
                                const float* __restrict__ p_tb,
                                const float* __restrict__ p_tw,
                                float* __restrict__ r_log,
                                float* __restrict__ t_log,
                                unsigned* __restrict__ m_r,
                                unsigned* __restrict__ m_t) {
  int e = blockIdx.x * blockDim.x + threadIdx.x;
  if (e >= N_EDGES) return;
  int s = ei[e];
  int r = ei[N_EDGES + e];
  float len = elen[e];
  float scale = softplusf(p_ls[0]);

  float4 srs = *(const float4*)(s_r + s * NH);
  float4 srr = *(const float4*)(s_r + r * NH);
  float4 sts = *(const float4*)(s_t + s * NH);
  float4 str = *(const float4*)(s_t + r * NH);

  float rl[NH], tl[NH];
  rl[0] = srs.x - srr.x; rl[1] = srs.y - srr.y; rl[2] = srs.z - srr.z; rl[3] = srs.w - srr.w;
  tl[0] = sts.x - str.x; tl[1] = sts.y - str.y; tl[2] = sts.z - str.z; tl[3] = sts.w - str.w;

#pragma unroll
  for (int h = 0; h < NH; ++h) {
    float temp = softplusf(p_tb[h] + p_tw[h] * len) + 0.0001f;
    rl[h] = (rl[h] - scale * len) / temp;
    r_log[e * NH + h] = rl[h];
    t_log[e * NH + h] = tl[h];
    atomicMax(&m_r[r * NH + h], fenc(rl[h]));
    atomicMax(&m_t[r * NH + h], fenc(tl[h]));
  }
}

// ------------------------------------------------- exp + segment sum
__global__ void ega_edge_exp(const int* __restrict__ ei,
                             float* __restrict__ r_log,  // in: logits, out: exp
                             float* __restrict__ t_log,
                             const unsigned* __restrict__ m_r,
                             const unsigned* __restrict__ m_t,
                             float* __restrict__ d_r,
                             float* __restrict__ d_t) {
  int e = blockIdx.x * blockDim.x + threadIdx.x;
  if (e >= N_EDGES) return;
  int r = ei[N_EDGES + e];
#pragma unroll
  for (int h = 0; h < NH; ++h) {
    float ar = __expf(r_log[e * NH + h] - fdec(m_r[r * NH + h]));
    float at = __expf(t_log[e * NH + h] - fdec(m_t[r * NH + h]));
    r_log[e * NH + h] = ar;
    t_log[e * NH + h] = at;
    atomicAdd(&d_r[r * NH + h], ar);
    atomicAdd(&d_t[r * NH + h], at);
  }
}

// ------------------------------------------------- message scatter (sender side only)
// Softmax weights sum to 1 per (receiver, head), so the receiver-side term
//   Σ_e w[e,h] * tab[h][r]  ==  tab[h][r]
// is applied analytically in ega_node_correct. Here we only gather sender rows.
// one wave per edge; lane handles feature pair (2*lane, 2*lane+1)
__global__ void ega_scatter(const int* __restrict__ ei,
                            const float* __restrict__ a_r,
                            const float* __restrict__ a_t,
                            const float* __restrict__ d_r,
                            const float* __restrict__ d_t,
                            const float* __restrict__ radial_tab,
                            const float* __restrict__ tang_tab,
                            float* __restrict__ msg) {
  int e = (blockIdx.x * blockDim.x + threadIdx.x) >> 5;
  if (e >= N_EDGES) return;
  int lane = threadIdx.x & 31;
  int s = ei[e];
  int r = ei[N_EDGES + e];
  int f0 = lane * 2;

  float accx = 0.f, accy = 0.f;
#pragma unroll
  for (int h = 0; h < NH; ++h) {
    float wr = a_r[e * NH + h] / d_r[r * NH + h];
    float wt = a_t[e * NH + h] / d_t[r * NH + h];
    const float* rbase = radial_tab + (size_t)h * N_NODES * FDIM;
    const float* tbase = tang_tab + (size_t)h * N_NODES * FDIM;
    float2 rsv = *(const float2*)(rbase + (size_t)s * FDIM + f0);
    float2 tsv = *(const float2*)(tbase + (size_t)s * FDIM + f0);
    accx += wr * rsv.x + wt * tsv.x;
    accy += wr * rsv.y + wt * tsv.y;
  }
  atomicAdd(&msg[(size_t)r * FDIM + f0], accx);
  atomicAdd(&msg[(size_t)r * FDIM + f0 + 1], accy);
}

// ------------------------------------------------- receiver-side correction
// msg[n][f] -= Σ_h (radial[h][n][f] + tang[h][n][f])   iff node n has >=1 incoming
// edge (d_r > 0 exactly then, since exp-sums are strictly positive). Isolated
// nodes keep msg == 0, matching the reference's segment_sum semantics.
__global__ void ega_node_correct(const float* __restrict__ d_r,
                                 const float* __restrict__ radial_tab,
                                 const float* __restrict__ tang_tab,
                                 float* __restrict__ msg) {
  int i = blockIdx.x * blockDim.x + threadIdx.x;
  if (i >= N_NODES * FDIM) return;
  int n = i >> 6;
  if (d_r[n * NH] <= 0.0f) return;  // isolated receiver: weights sum to 0, not 1
  float sub = 0.f;
#pragma unroll
  for (int h = 0; h < NH; ++h) {
    sub += radial_tab[(size_t)h * N_NODES * FDIM + i];
    sub += tang_tab[(size_t)h * N_NODES * FDIM + i];
  }
  msg[i] -= sub;
}

// ------------------------------------------------- output GEMM (WMMA) + residual
// out = x + (msg / NH) @ w_out
__global__ void ega_out_wmma(const float* __restrict__ msg,
                             const float* __restrict__ x,
                             const float* __restrict__ wout,
                             float* __restrict__ out) {
  int wave = (blockIdx.x * blockDim.x + threadIdx.x) >> 5;
  if (wave >= MTILES) return;
  int lane = threadIdx.x & 31;
  int half = lane >> 4;
  int lm   = lane & 15;
  int m0   = wave * 16;

  const float invH = 1.0f / (float)NH;
  v2f areg[16];
  const float* mrow = msg + (size_t)(m0 + lm) * FDIM + half * 2;
#pragma unroll
  for (int k4 = 0; k4 < 16; ++k4) {
    v2f a = *(const v2f*)(mrow + k4 * 4);
    a.x *= invH;
    a.y *= invH;
    areg[k4] = a;
  }

  v8f acc[4];
#pragma unroll
  for (int t = 0; t < 4; ++t) acc[t] = (v8f){0.f, 0.f, 0.f, 0.f, 0.f, 0.f, 0.f, 0.f};

#pragma unroll
  for (int t = 0; t < 4; ++t) {
#pragma unroll
    for (int k4 = 0; k4 < 16; ++k4) {
      int kk = k4 * 4 + half * 2;
      v2f b;
      b.x = wout[kk * FDIM + t * 16 + lm];
      b.y = wout[(kk + 1) * FDIM + t * 16 + lm];
      acc[t] = __builtin_amdgcn_wmma_f32_16x16x4_f32(
          false, areg[k4], false, b, (short)0, acc[t], false, false);
    }
  }

#pragma unroll
  for (int t = 0; t < 4; ++t) {
#pragma unroll
    for (int v = 0; v < 8; ++v) {
      int row = m0 + v + half * 8;
      size_t idx = (size_t)row * FDIM + t * 16 + lm;
      out[idx] = x[idx] + acc[t][v];
    }
  }
}

// ---------------------------------------------------------------- launch
extern "C" void kernel_launch(void* const* d_in, const int* in_sizes, int n_in,
                              void* d_out, int out_size, void* d_ws, size_t ws_size,
                              hipStream_t stream) {
  const float* x    = (const float*)d_in[0];
  const int*   ei   = (const int*)d_in[1];
  // d_in[2] = edge_vec (unused by the reference)
  const float* elen = (const float*)d_in[3];
  const float* wp   = (const float*)d_in[4];
  const float* wr   = (const float*)d_in[5];
  const float* wt   = (const float*)d_in[6];
  const float* rs   = (const float*)d_in[7];
  const float* ts   = (const float*)d_in[8];
  const float* ls   = (const float*)d_in[9];
  const float* tb   = (const float*)d_in[10];
  const float* tw   = (const float*)d_in[11];
  const float* wout = (const float*)d_in[12];
  float* out = (float*)d_out;

  // workspace layout (float elements)
  float* ws = (float*)d_ws;
  float*    radial_tab = ws;                                     // 2,560,000
  float*    tang_tab   = ws + (size_t)NH * N_NODES * FDIM;       // 2,560,000
  float*    s_r        = tang_tab + (size_t)NH * N_NODES * FDIM; // 40,000
  float*    s_t        = s_r + N_NODES * NH;                     // 40,000
  float*    r_log      = s_t + N_NODES * NH;                     // 640,000
  float*    t_log      = r_log + N_EDGES * NH;                   // 640,000
  unsigned* m_r        = (unsigned*)(t_log + N_EDGES * NH);      // 40,000
  unsigned* m_t        = m_r + N_NODES * NH;                     // 40,000
  float*    d_r        = (float*)(m_t + N_NODES * NH);           // 40,000
  float*    d_t        = d_r + N_NODES * NH;                     // 40,000
  float*    msg        = d_t + N_NODES * NH;                     // 640,000

  ega_init<<<(N_NODES * FDIM + 255) / 256, 256, 0, stream>>>(m_r, m_t, d_r, d_t, msg);

  int proj_waves = 3 * NH * MTILES;                 // 7500 waves
  ega_proj_wmma<<<(proj_waves * 32 + 255) / 256, 256, 0, stream>>>(
      x, wp, wr, wt, rs, ts, radial_tab, tang_tab, s_r, s_t);

  ega_edge_logits<<<(N_EDGES + 255) / 256, 256, 0, stream>>>(
      ei, elen, s_r, s_t, ls, tb, tw, r_log, t_log, m_r, m_t);

  ega_edge_exp<<<(N_EDGES + 255) / 256, 256, 0, stream>>>(
      ei, r_log, t_log, m_r, m_t, d_r, d_t);

  ega_scatter<<<(N_EDGES * 32 + 255) / 256, 256, 0, stream>>>(
      ei, r_log, t_log, d_r, d_t, radial_tab, tang_tab, msg);

  ega_node_correct<<<(N_NODES * FDIM + 255) / 256, 256, 0, stream>>>(
      d_r, radial_tab, tang_tab, msg);

  ega_out_wmma<<<(MTILES * 32 + 255) / 256, 256, 0, stream>>>(msg, x, wout, out);
}